// TinyRecursive_80590766342523
// MI455X (gfx1250) — compile-verified
//
#include <hip/hip_runtime.h>

typedef __attribute__((ext_vector_type(2))) float v2f;
typedef __attribute__((ext_vector_type(8))) float v8f;

#define LDS_STRIDE 260   // floats; 260 % 64 == 4 -> conflict-free b64 A-frag loads
#define D_IN 256

// D = A(16x4 f32) x B(4x16 f32) + C(16x16 f32)
__device__ __forceinline__ v8f wmma4(v2f a, v2f b, v8f c) {
  return __builtin_amdgcn_wmma_f32_16x16x4_f32(
      /*neg_a=*/false, a, /*neg_b=*/false, b,
      /*c_mod=*/(short)0, c, /*reuse_a=*/false, /*reuse_b=*/false);
}

// Build one K-chunk B-fragment (rows r0..r3 of the conceptual KxN matrix) from
// four C-layout rows whose valid data lives in lanes 0-15.
// vgpr j: lanes 0-15 = row r0+j (col = lane), lanes 16-31 = row r0+2+j (col = lane-16)
__device__ __forceinline__ v2f make_bfrag(float r0, float r1, float r2, float r3, int lane) {
  float t2 = __shfl(r2, lane & 15, 32);
  float t3 = __shfl(r3, lane & 15, 32);
  v2f b;
  b.x = (lane < 16) ? r0 : t2;
  b.y = (lane < 16) ? r1 : t3;
  return b;
}

__global__ __launch_bounds__(256) void tinyrec_wmma_kernel(
    const float* __restrict__ x,     const float* __restrict__ w_in,
    const float* __restrict__ b_in,  const float* __restrict__ w1,
    const float* __restrict__ b1,    const float* __restrict__ w2,
    const float* __restrict__ b2,    const float* __restrict__ gamma,
    const float* __restrict__ beta,  const float* __restrict__ w_out,
    const float* __restrict__ b_out, const int* __restrict__ Kptr,
    float* __restrict__ out, int nRows)
{
  __shared__ float Wlds[16 * LDS_STRIDE];

  const int tid  = threadIdx.x;
  const int lane = tid & 31;
  const int wave = tid >> 5;
  const int m    = lane & 15;   // row-in-tile / M index
  const int half = lane >> 4;   // lane half selects K pair

  // Stage w_in^T (padded 16 x 256, zeros for m>=8) into LDS, stride 260.
  for (int i = tid; i < 16 * D_IN; i += 256) {
    int mm = i >> 8;
    int k  = i & 255;
    Wlds[mm * LDS_STRIDE + k] = (mm < 8) ? w_in[k * 8 + mm] : 0.0f;
  }
  __syncthreads();

  const long tileRow = (((long)blockIdx.x) * 8 + wave) * 16;
  if (tileRow >= nRows) return;

  // Uniform parameters -> scalar regs
  float s_bin[8], s_b1[8], s_b2[8], s_g[8], s_bt[8];
#pragma unroll
  for (int g = 0; g < 8; ++g) {
    s_bin[g] = b_in[g]; s_b1[g] = b1[g]; s_b2[g] = b2[g];
    s_g[g] = gamma[g];  s_bt[g] = beta[g];
  }

  // Constant A-fragments: w1^T (16x16, 4 K-chunks) and w2^T (16x8, 2 K-chunks).
  // A[m][k] = w{1,2}[k][m] for m<8, zero-padded rows m>=8.
  v2f a1[4], a2[2];
#pragma unroll
  for (int kc = 0; kc < 4; ++kc) {
    int kb = 4 * kc + 2 * half;
    float u0 = 0.0f, u1 = 0.0f;
    if (m < 8) { u0 = w1[kb * 8 + m]; u1 = w1[(kb + 1) * 8 + m]; }
    a1[kc].x = u0; a1[kc].y = u1;
  }
#pragma unroll
  for (int kc = 0; kc < 2; ++kc) {
    int kb = 4 * kc + 2 * half;
    float u0 = 0.0f, u1 = 0.0f;
    if (m < 8) { u0 = w2[kb * 8 + m]; u1 = w2[(kb + 1) * 8 + m]; }
    a2[kc].x = u0; a2[kc].y = u1;
  }

  // ---------------- Phase 1: x_proj^T = w_in^T @ x^T ----------------
  // B-frag: lane holds x[tileRow + m][4kc + 2*half + {0,1}]  (n = m, k per layout)
  const float* xrow = x + (tileRow + m) * (long)D_IN + 2 * half;
  v8f acc0 = {}, acc1 = {};
#pragma unroll 8
  for (int kc = 0; kc < 64; kc += 2) {
    v2f xb0 = __builtin_nontemporal_load((const v2f*)(xrow + 4 * kc));
    v2f xb1 = __builtin_nontemporal_load((const v2f*)(xrow + 4 * (kc + 1)));
    v2f aw0 = *(const v2f*)(&Wlds[m * LDS_STRIDE + 4 * kc + 2 * half]);
    v2f aw1 = *(const v2f*)(&Wlds[m * LDS_STRIDE + 4 * (kc + 1) + 2 * half]);
    acc0 = wmma4(aw0, xb0, acc0);
    acc1 = wmma4(aw1, xb1, acc1);
  }

  // x_proj^T rows 0-7 (+ bias); rows 8-15 are zero (A padding).
  float xp[8];
#pragma unroll
  for (int g = 0; g < 8; ++g) xp[g] = (acc0[g] + acc1[g]) + s_bin[g];

  // h^T rows 8-15 (= x_proj^T rows 0-7) B-fragments are loop-invariant.
  v2f bX2 = make_bfrag(xp[0], xp[1], xp[2], xp[3], lane);
  v2f bX3 = make_bfrag(xp[4], xp[5], xp[6], xp[7], lane);

  // ---------------- Phase 2: K recursion ----------------
  float z[8];
#pragma unroll
  for (int g = 0; g < 8; ++g) z[g] = 0.0f;

  const int K = *Kptr;
  for (int it = 0; it < K; ++it) {
    // h^T rows 0-7 = z^T
    v2f bZ0 = make_bfrag(z[0], z[1], z[2], z[3], lane);
    v2f bZ1 = make_bfrag(z[4], z[5], z[6], z[7], lane);

    // u^T = w1^T @ h^T   (4 WMMAs over K=16)
    v8f u = {};
    u = wmma4(a1[0], bZ0, u);
    u = wmma4(a1[1], bZ1, u);
    u = wmma4(a1[2], bX2, u);
    u = wmma4(a1[3], bX3, u);

    // relu(u + b1)
    float r[8];
#pragma unroll
    for (int g = 0; g < 8; ++g) r[g] = fmaxf(u[g] + s_b1[g], 0.0f);

    v2f bU0 = make_bfrag(r[0], r[1], r[2], r[3], lane);
    v2f bU1 = make_bfrag(r[4], r[5], r[6], r[7], lane);

    // v^T = w2^T @ relu^T + z   (seed C with z -> z + u directly)
    v8f v = {};
#pragma unroll
    for (int g = 0; g < 8; ++g) v[g] = z[g];
    v = wmma4(a2[0], bU0, v);
    v = wmma4(a2[1], bU1, v);

    // LayerNorm over H=8: features live along VGPR index -> per-lane reduction.
    float t[8], sum = 0.0f;
#pragma unroll
    for (int g = 0; g < 8; ++g) { t[g] = v[g] + s_b2[g]; sum += t[g]; }
    float mu = sum * 0.125f;
    float va = 0.0f;
#pragma unroll
    for (int g = 0; g < 8; ++g) { float d = t[g] - mu; t[g] = d; va += d * d; }
    float rs = __frsqrt_rn(va * 0.125f + 1e-5f);
#pragma unroll
    for (int g = 0; g < 8; ++g) z[g] = t[g] * rs * s_g[g] + s_bt[g];
  }

  // ---------------- Head: out = z @ w_out + b_out ----------------
  float wo[16];
#pragma unroll
  for (int i = 0; i < 16; ++i) wo[i] = w_out[i];
  float o0 = b_out[0], o1 = b_out[1];
#pragma unroll
  for (int g = 0; g < 8; ++g) { o0 += z[g] * wo[2 * g]; o1 += z[g] * wo[2 * g + 1]; }

  if (lane < 16) {
    v2f o; o.x = o0; o.y = o1;
    *(v2f*)(out + (tileRow + m) * 2) = o;
  }
}

extern "C" void kernel_launch(void* const* d_in, const int* in_sizes, int n_in,
                              void* d_out, int out_size, void* d_ws, size_t ws_size,
                              hipStream_t stream) {
  const float* x     = (const float*)d_in[0];
  const float* w_in  = (const float*)d_in[1];
  const float* b_in  = (const float*)d_in[2];
  const float* w1    = (const float*)d_in[3];
  const float* b1    = (const float*)d_in[4];
  const float* w2    = (const float*)d_in[5];
  const float* b2    = (const float*)d_in[6];
  const float* gamma = (const float*)d_in[7];
  const float* beta  = (const float*)d_in[8];
  const float* w_out = (const float*)d_in[9];
  const float* b_out = (const float*)d_in[10];
  const int*   Kp    = (const int*)d_in[11];

  int nRows  = in_sizes[0] / D_IN;          // B
  int tiles  = (nRows + 15) / 16;
  int blocks = (tiles + 7) / 8;             // 8 waves (tiles) per 256-thread block

  tinyrec_wmma_kernel<<<blocks, 256, 0, stream>>>(
      x, w_in, b_in, w1, b1, w2, b2, gamma, beta, w_out, b_out, Kp,
      (float*)d_out, nRows);
}